// SelfAttention_24524263260918
// MI455X (gfx1250) — compile-verified
//
#include <hip/hip_runtime.h>
#include <math.h>

// ---------------------------------------------------------------------------
// CDNA5 (gfx1250, wave32) self-attention, full-f32 via V_WMMA_F32_16X16X4_F32.
// Fused scores+mask+softmax+AV kernel keeps the 16x1024 score strip in LDS,
// eliminating 768 MB of HBM round-trips on the attention tensor.
// ---------------------------------------------------------------------------

typedef float v2f __attribute__((ext_vector_type(2)));
typedef float v8f __attribute__((ext_vector_type(8)));

#define BATCH   4
#define L_SEQ   1024
#define DMODEL  1024
#define HEADS   16
#define DHEAD   64
#define MIN_VALUE (-1.7014118e38f)   /* finfo(f32).min / 2 */
#define LN_EPS    1e-5f
#define S_STRIDE  1028               /* padded LDS row stride (words), %64==4 */

static __device__ __forceinline__ v8f wmma_f32(v2f a, v2f b, v8f c) {
  // 8 args: (neg_a, A, neg_b, B, c_mod, C, reuse_a, reuse_b)
  return __builtin_amdgcn_wmma_f32_16x16x4_f32(false, a, false, b, (short)0, c,
                                               false, false);
}

// A fragment: A row-major, rows row0..row0+15, K cols k0..k0+3.
// lanes 0-15 -> K = {k0, k0+1}; lanes 16-31 -> K = {k0+2, k0+3}
static __device__ __forceinline__ v2f load_a_frag(const float* __restrict__ A,
                                                  int lda, int row0, int k0,
                                                  int lane) {
  int half = lane >> 4;
  int r    = row0 + (lane & 15);
  const float* p = A + (size_t)r * lda + k0 + 2 * half;
  return *(const v2f*)p;  // lda even, k0 % 4 == 0 -> 8B aligned
}

// B fragment from row-major B[K x ldb]
static __device__ __forceinline__ v2f load_b_row(const float* __restrict__ B,
                                                 int ldb, int k0, int col0,
                                                 int lane) {
  int half = lane >> 4;
  int n    = col0 + (lane & 15);
  const float* p = B + (size_t)(k0 + 2 * half) * ldb + n;
  v2f b;
  b[0] = p[0];
  b[1] = p[ldb];
  return b;
}

// B fragment where B = M^T, M row-major [N x ldm]  (B[k][n] = M[n][k])
static __device__ __forceinline__ v2f load_b_trans(const float* __restrict__ M,
                                                   int ldm, int k0, int col0,
                                                   int lane) {
  int half = lane >> 4;
  int n    = col0 + (lane & 15);
  const float* p = M + (size_t)n * ldm + k0 + 2 * half;
  return *(const v2f*)p;
}

// ---------------------------------------------------------------------------
// K1: q/k/v = x @ W[h] + b[h].  grid(L/64, B*H, 3), block(128) = 4 waves.
// Wave w computes 64 rows x 16 cols (col-tile w): B frag reused across the
// 4 row tiles in-register -> 6 vmem per 4 WMMA.
// ---------------------------------------------------------------------------
__global__ void qkv_proj_kernel(const float* __restrict__ x,
                                const float* __restrict__ Wq,
                                const float* __restrict__ bq,
                                const float* __restrict__ Wk,
                                const float* __restrict__ bk,
                                const float* __restrict__ Wv,
                                const float* __restrict__ bv,
                                float* __restrict__ q, float* __restrict__ k,
                                float* __restrict__ v) {
  const int wave = threadIdx.x >> 5, lane = threadIdx.x & 31;
  const int bh = blockIdx.y, b = bh >> 4, h = bh & 15;
  const int rt   = blockIdx.x * 64;
  const int col0 = wave * 16;

  const float* W;
  const float* bias;
  float* out;
  if (blockIdx.z == 0)      { W = Wq; bias = bq; out = q; }
  else if (blockIdx.z == 1) { W = Wk; bias = bk; out = k; }
  else                      { W = Wv; bias = bv; out = v; }

  const float* A  = x + (size_t)b * L_SEQ * DMODEL;
  const float* Bm = W + (size_t)h * DMODEL * DHEAD;

  v8f acc[4] = {};
#pragma unroll 2
  for (int k0 = 0; k0 < DMODEL; k0 += 4) {
    v2f bb = load_b_row(Bm, DHEAD, k0, col0, lane);
#pragma unroll
    for (int i = 0; i < 4; ++i) {
      v2f a = load_a_frag(A, DMODEL, rt + i * 16, k0, lane);
      acc[i] = wmma_f32(a, bb, acc[i]);
    }
  }

  const int n  = col0 + (lane & 15);
  const int rb = (lane >> 4) << 3;
  const float bias_n = bias[h * DHEAD + n];
  float* D = out + (size_t)bh * L_SEQ * DHEAD;
#pragma unroll
  for (int i = 0; i < 4; ++i)
#pragma unroll
    for (int r = 0; r < 8; ++r)
      D[(size_t)(rt + i * 16 + rb + r) * DHEAD + n] = acc[i][r] + bias_n;
}

// ---------------------------------------------------------------------------
// K2 (fused): scores -> mask -> softmax -> atten (HBM, once) -> atten @ V.
// grid(L/16, B*H), block(512) = 16 waves.  Wave w owns score cols
// [64w, 64w+64).  Score strip (16 x 1024) lives in LDS for the AV phase.
// ---------------------------------------------------------------------------
__global__ void __launch_bounds__(512)
attention_kernel(const float* __restrict__ q, const float* __restrict__ kmat,
                 const float* __restrict__ vmat, const int* __restrict__ mask,
                 float* __restrict__ atten, float* __restrict__ ctx) {
  __shared__ float s_lds[16 * S_STRIDE];  // q tile, then scores, then partials
  __shared__ float red[16 * 16];          // per-wave per-row partials
  __shared__ float bc[16];                // broadcast row max / row sum

  const int tid = threadIdx.x, wave = tid >> 5, lane = tid & 31;
  const int half = lane >> 4;
  const int bh = blockIdx.y, b = bh >> 4, h = bh & 15;
  const int row0 = blockIdx.x * 16;

  const float* Q = q    + (size_t)bh * L_SEQ * DHEAD;
  const float* K = kmat + (size_t)bh * L_SEQ * DHEAD;
  const float* V = vmat + (size_t)bh * L_SEQ * DHEAD;

  // ---- stage Q tile (16 x 64) into LDS with stride 68 (banks %64 == 4) ----
  for (int i = tid; i < 16 * DHEAD; i += 512) {
    int r = i >> 6, d = i & 63;
    s_lds[r * 68 + d] = Q[(size_t)(row0 + r) * DHEAD + d];
  }
  __syncthreads();

  // ---- phase 1: scores = Q @ K^T (wave w: cols 64w..64w+63) ----
  v8f acc[4] = {};
#pragma unroll
  for (int k0 = 0; k0 < DHEAD; k0 += 4) {
    v2f a = *(const v2f*)&s_lds[(lane & 15) * 68 + k0 + 2 * half];
#pragma unroll
    for (int t = 0; t < 4; ++t) {
      v2f bb = load_b_trans(K, DHEAD, k0, wave * 64 + t * 16, lane);
      acc[t] = wmma_f32(a, bb, acc[t]);
    }
  }

  // ---- scale + mask ----
  float sv[4][8];
#pragma unroll
  for (int t = 0; t < 4; ++t) {
    const bool msk = (mask[b * L_SEQ + wave * 64 + t * 16 + (lane & 15)] == 0);
#pragma unroll
    for (int r = 0; r < 8; ++r)
      sv[t][r] = msk ? MIN_VALUE : acc[t][r] * 0.125f;  // 1/sqrt(64)
  }

  // ---- phase 2: row max (lanes-in-half -> waves) ----
  float pm[8];
#pragma unroll
  for (int r = 0; r < 8; ++r) {
    pm[r] = fmaxf(fmaxf(sv[0][r], sv[1][r]), fmaxf(sv[2][r], sv[3][r]));
#pragma unroll
    for (int o = 8; o > 0; o >>= 1) pm[r] = fmaxf(pm[r], __shfl_xor(pm[r], o, 32));
  }
  if ((lane & 15) == 0)
#pragma unroll
    for (int r = 0; r < 8; ++r) red[wave * 16 + half * 8 + r] = pm[r];
  __syncthreads();
  if (tid < 16) {
    float m = red[tid];
#pragma unroll
    for (int w = 1; w < 16; ++w) m = fmaxf(m, red[w * 16 + tid]);
    bc[tid] = m;
  }
  __syncthreads();
  float rmax[8];
#pragma unroll
  for (int r = 0; r < 8; ++r) rmax[r] = bc[half * 8 + r];

  // ---- phase 3: exp + row sum ----
  float ps[8];
#pragma unroll
  for (int r = 0; r < 8; ++r) ps[r] = 0.f;
#pragma unroll
  for (int t = 0; t < 4; ++t)
#pragma unroll
    for (int r = 0; r < 8; ++r) {
      float e = expf(sv[t][r] - rmax[r]);
      sv[t][r] = e;
      ps[r] += e;
    }
#pragma unroll
  for (int r = 0; r < 8; ++r)
#pragma unroll
    for (int o = 8; o > 0; o >>= 1) ps[r] += __shfl_xor(ps[r], o, 32);
  if ((lane & 15) == 0)
#pragma unroll
    for (int r = 0; r < 8; ++r) red[wave * 16 + half * 8 + r] = ps[r];
  __syncthreads();
  if (tid < 16) {
    float s = 0.f;
#pragma unroll
    for (int w = 0; w < 16; ++w) s += red[w * 16 + tid];
    bc[tid] = s;
  }
  __syncthreads();
  float rinv[8];
#pragma unroll
  for (int r = 0; r < 8; ++r) rinv[r] = 1.0f / bc[half * 8 + r];
  __syncthreads();  // everyone done with q region of s_lds before overwrite

  // ---- phase 4: write atten (HBM, once) and stage scores in LDS ----
  float* D = atten + (size_t)bh * L_SEQ * L_SEQ;
#pragma unroll
  for (int t = 0; t < 4; ++t) {
    const int n = wave * 64 + t * 16 + (lane & 15);
#pragma unroll
    for (int r = 0; r < 8; ++r) {
      const int rr = half * 8 + r;
      const float val = sv[t][r] * rinv[r];
      D[(size_t)(row0 + rr) * L_SEQ + n] = val;
      s_lds[rr * S_STRIDE + n] = val;
    }
  }
  __syncthreads();

  // ---- phase 5: partial AV.  Wave w: K slice [64w, 64w+64). ----
  v8f pacc[4] = {};
  const int kbase = wave * 64;
#pragma unroll
  for (int kk = 0; kk < 64; kk += 4) {
    const int k0 = kbase + kk;
    v2f a = *(const v2f*)&s_lds[(lane & 15) * S_STRIDE + k0 + 2 * half];
#pragma unroll
    for (int t = 0; t < 4; ++t) {
      v2f bb = load_b_row(V, DHEAD, k0, t * 16, lane);
      pacc[t] = wmma_f32(a, bb, pacc[t]);
    }
  }
  __syncthreads();  // all waves done reading scores before overwrite

  // partials -> LDS [wave][16][64]
#pragma unroll
  for (int t = 0; t < 4; ++t)
#pragma unroll
    for (int r = 0; r < 8; ++r) {
      const int rr = half * 8 + r;
      s_lds[wave * 1024 + rr * 64 + t * 16 + (lane & 15)] = pacc[t][r];
    }
  __syncthreads();

  // tree-reduce 16 partials, write ctx[b, row0+rr, h*64 + d]
  for (int e = tid; e < 1024; e += 512) {
    float s = 0.f;
#pragma unroll
    for (int w = 0; w < 16; ++w) s += s_lds[w * 1024 + e];
    const int rr = e >> 6, d = e & 63;
    ctx[((size_t)b * L_SEQ + row0 + rr) * DMODEL + h * DHEAD + d] = s;
  }
}

// ---------------------------------------------------------------------------
// K3a: pre = ctx @ Wo^T + bo + x.  grid(DMODEL/64, B*L/64), block(128).
// Wave w: 64 rows x 16 cols (col-tile w), Wo^T frag reused in-register.
// ---------------------------------------------------------------------------
__global__ void oproj_kernel(const float* __restrict__ ctx,
                             const float* __restrict__ Wo,
                             const float* __restrict__ bo,
                             const float* __restrict__ x,
                             float* __restrict__ pre) {
  const int wave = threadIdx.x >> 5, lane = threadIdx.x & 31;
  const int rt   = blockIdx.y * 64;
  const int col0 = blockIdx.x * 64 + wave * 16;

  v8f acc[4] = {};
#pragma unroll 2
  for (int k0 = 0; k0 < DMODEL; k0 += 4) {
    v2f bb = load_b_trans(Wo, DMODEL, k0, col0, lane);  // B = Wo^T
#pragma unroll
    for (int i = 0; i < 4; ++i) {
      v2f a = load_a_frag(ctx, DMODEL, rt + i * 16, k0, lane);
      acc[i] = wmma_f32(a, bb, acc[i]);
    }
  }

  const int n  = col0 + (lane & 15);
  const int rb = (lane >> 4) << 3;
  const float bias_n = bo[n];
#pragma unroll
  for (int i = 0; i < 4; ++i)
#pragma unroll
    for (int r = 0; r < 8; ++r) {
      size_t idx = (size_t)(rt + i * 16 + rb + r) * DMODEL + n;
      pre[idx] = acc[i][r] + bias_n + x[idx];
    }
}

// ---------------------------------------------------------------------------
// reductions (wave32) + K3b LayerNorm.  grid(B*L), block(256).
// ---------------------------------------------------------------------------
static __device__ __forceinline__ float wave_sum(float v) {
#pragma unroll
  for (int o = 16; o > 0; o >>= 1) v += __shfl_xor(v, o, 32);
  return v;
}

__global__ void ln_kernel(const float* __restrict__ pre,
                          const float* __restrict__ gamma,
                          const float* __restrict__ beta,
                          float* __restrict__ out) {
  __shared__ float sred[8];
  __shared__ float sbc;
  const int tid = threadIdx.x, lane = tid & 31, wid = tid >> 5;
  const float* p = pre + (size_t)blockIdx.x * DMODEL;

  float vals[4];
  float s = 0.f;
#pragma unroll
  for (int j = 0; j < 4; ++j) {
    vals[j] = p[tid + j * 256];
    s += vals[j];
  }
  s = wave_sum(s);
  if (lane == 0) sred[wid] = s;
  __syncthreads();
  if (tid == 0) {
    float t = 0.f;
#pragma unroll
    for (int i = 0; i < 8; ++i) t += sred[i];
    sbc = t;
  }
  __syncthreads();
  const float mu = sbc * (1.0f / DMODEL);
  __syncthreads();

  float sq = 0.f;
#pragma unroll
  for (int j = 0; j < 4; ++j) {
    float d = vals[j] - mu;
    sq += d * d;
  }
  sq = wave_sum(sq);
  if (lane == 0) sred[wid] = sq;
  __syncthreads();
  if (tid == 0) {
    float t = 0.f;
#pragma unroll
    for (int i = 0; i < 8; ++i) t += sred[i];
    sbc = t;
  }
  __syncthreads();
  const float rs = rsqrtf(sbc * (1.0f / DMODEL) + LN_EPS);
  float* o = out + (size_t)blockIdx.x * DMODEL;
#pragma unroll
  for (int j = 0; j < 4; ++j) {
    int n = tid + j * 256;
    o[n] = (vals[j] - mu) * rs * gamma[n] + beta[n];
  }
}

// ---------------------------------------------------------------------------
// Launcher.  Inputs: x, mask, Wq, bq, Wk, bk, Wv, bv, Wo, bo, gamma, beta.
// d_out = [ out: 4*1024*1024 f32 | atten: 4*16*1024*1024 f32 ].
// Workspace (floats): q|k|v (3 * 4,194,304) + ctx (4,194,304) = 64 MB;
// pre-LN buffer aliases q (dead after the fused attention kernel).
// ---------------------------------------------------------------------------
extern "C" void kernel_launch(void* const* d_in, const int* in_sizes, int n_in,
                              void* d_out, int out_size, void* d_ws,
                              size_t ws_size, hipStream_t stream) {
  const float* x     = (const float*)d_in[0];
  const int*   mask  = (const int*)d_in[1];
  const float* Wq    = (const float*)d_in[2];
  const float* bq    = (const float*)d_in[3];
  const float* Wk    = (const float*)d_in[4];
  const float* bk    = (const float*)d_in[5];
  const float* Wv    = (const float*)d_in[6];
  const float* bv    = (const float*)d_in[7];
  const float* Wo    = (const float*)d_in[8];
  const float* bo    = (const float*)d_in[9];
  const float* gamma = (const float*)d_in[10];
  const float* beta  = (const float*)d_in[11];

  float* out_ln = (float*)d_out;                                   // [B,L,1024]
  float* atten  = (float*)d_out + (size_t)BATCH * L_SEQ * DMODEL;  // [B,H,L,L]

  float* ws = (float*)d_ws;
  const size_t perQKV = (size_t)BATCH * HEADS * L_SEQ * DHEAD;  // 4,194,304
  float* q   = ws;
  float* k   = ws + perQKV;
  float* v   = ws + 2 * perQKV;
  float* ctx = ws + 3 * perQKV;  // [B, L, DMODEL]
  float* pre = ws;               // aliases q (dead after attention kernel)

  dim3 blk128(128), blk256(256), blk512(512);

  // 1) QKV projections
  qkv_proj_kernel<<<dim3(L_SEQ / 64, BATCH * HEADS, 3), blk128, 0, stream>>>(
      x, Wq, bq, Wk, bk, Wv, bv, q, k, v);

  // 2) fused scores + mask + softmax + atten write + atten @ V
  attention_kernel<<<dim3(L_SEQ / 16, BATCH * HEADS), blk512, 0, stream>>>(
      q, k, v, mask, atten, ctx);

  // 3a) output projection + bias + residual
  oproj_kernel<<<dim3(DMODEL / 64, BATCH * L_SEQ / 64), blk128, 0, stream>>>(
      ctx, Wo, bo, x, pre);

  // 3b) LayerNorm
  ln_kernel<<<dim3(BATCH * L_SEQ), blk256, 0, stream>>>(pre, gamma, beta,
                                                        out_ln);
}